// WSegPC_Loss_7851200217678
// MI455X (gfx1250) — compile-verified
//
#include <hip/hip_runtime.h>
#include <math.h>

#define B_  4
#define V_  4
#define C_  20
#define H_  240
#define W_  320
#define HW_ (H_*W_)
#define N_  2000000
#define P_  100000
#define D_  64
#define K_  1024
#define TAU_ 0.07f
#define NEG_ (-1000000.0f)

typedef __attribute__((ext_vector_type(2))) float v2f;
typedef __attribute__((ext_vector_type(8))) float v8f;
typedef int aint4 __attribute__((vector_size(4 * sizeof(int))));

// ---- CDNA5 async global->LDS path (guarded; falls back to ld+ds_store) ----
#if defined(__has_builtin)
#if __has_builtin(__builtin_amdgcn_global_load_async_to_lds_b128)
#define ASYNC_LDS 1
#endif
#endif

#if defined(ASYNC_LDS)
#define G1(p) ((__attribute__((address_space(1))) aint4*)(p))
#define L3(p) ((__attribute__((address_space(3))) aint4*)(p))
#if __has_builtin(__builtin_amdgcn_s_wait_asynccnt)
#define ASYNC_WAIT(n) __builtin_amdgcn_s_wait_asynccnt(n)
#else
#define ASYNC_WAIT(n) asm volatile("s_wait_asynccnt %0" :: "i"(n) : "memory")
#endif
#else
#define ASYNC_WAIT(n) ((void)0)
#endif

// ---------------------------------------------------------------- init ----
__global__ void init_kernel(float* __restrict__ ws, float* __restrict__ out) {
    int tid = threadIdx.x;
    if (tid < 164) ws[tid] = 0.0f;   // [0..79] seg_sum, [80..83] cnt, [84..163] predict_sum
    if (tid == 0)  out[0] = 0.0f;    // loss scalar
}

// ------------------------------------------------------------- 3D CAM ----
__global__ __launch_bounds__(256) void cam3d_kernel(
    const float* __restrict__ cam3d, const int* __restrict__ invmap,
    const int* __restrict__ bidx,    const float* __restrict__ lab3,
    float* __restrict__ out3, float* __restrict__ ws)
{
    __shared__ float slab[B_*C_];
    __shared__ float ssum[B_*C_];
    __shared__ float scnt[B_];
    __shared__ float ostage[256 * C_];           // 20KB staged output tile
    int tid = threadIdx.x;
    if (tid < B_*C_) { slab[tid] = lab3[tid]; ssum[tid] = 0.0f; }
    if (tid < B_)    scnt[tid] = 0.0f;
    __syncthreads();

    int n0 = blockIdx.x * 256;
    int n  = n0 + tid;
    if (n < N_) {
        int b = bidx[n];
        size_t row = (size_t)invmap[n];
        float x[C_], m[C_];
        const float4* sp = (const float4*)(cam3d + row * C_);   // rows 80B, 16B aligned
        #pragma unroll
        for (int i = 0; i < 5; ++i) {
            float4 v = sp[i];
            x[4*i+0]=v.x; x[4*i+1]=v.y; x[4*i+2]=v.z; x[4*i+3]=v.w;
        }
        float mx = -INFINITY;
        #pragma unroll
        for (int c = 0; c < C_; ++c) {
            m[c] = x[c] + (1.0f - slab[b*C_+c]) * NEG_;
            mx = fmaxf(mx, m[c]);
        }
        float se = 0.0f;
        #pragma unroll
        for (int c = 0; c < C_; ++c) { m[c] = expf(m[c] - mx); se += m[c]; }
        float inv = 1.0f / se;
        #pragma unroll
        for (int c = 0; c < C_; ++c) ostage[tid*C_ + c] = m[c] * inv;
        #pragma unroll
        for (int c = 0; c < C_; ++c) atomicAdd(&ssum[b*C_+c], x[c]);   // raw values
        atomicAdd(&scnt[b], 1.0f);
    }
    __syncthreads();
    // coalesced writeback of the block's contiguous output region
    {
        size_t base = (size_t)n0 * C_;
        int valid = (N_ - n0 < 256 ? N_ - n0 : 256) * C_;
        #pragma unroll
        for (int i = 0; i < C_; ++i) {
            int o = i * 256 + tid;
            if (o < valid) out3[base + o] = ostage[o];
        }
    }
    if (tid < B_*C_) atomicAdd(&ws[tid], ssum[tid]);
    if (tid < B_)    atomicAdd(&ws[80+tid], scnt[tid]);
}

// ------------------------------------------------------------- 2D CAM ----
__global__ __launch_bounds__(256) void cam2d_kernel(
    const float* __restrict__ cam2d, const float* __restrict__ lab2,
    float* __restrict__ out2, float* __restrict__ ws)
{
    __shared__ float slab[B_*C_];
    __shared__ float psum[B_*C_];
    int tid = threadIdx.x;
    if (tid < B_*C_) { slab[tid] = lab2[tid]; psum[tid] = 0.0f; }
    __syncthreads();

    int p  = blockIdx.x * 256 + tid;        // p < B*V*HW = 1,228,800 exactly
    int hw = p % HW_;
    int bv = p / HW_;
    int b  = bv / V_;
    int v  = bv % V_;

    const float* ip = cam2d + (size_t)bv * C_ * HW_ + hw;
    float x[C_], m[C_];
    float mx = -INFINITY;
    #pragma unroll
    for (int c = 0; c < C_; ++c) {
        x[c] = ip[(size_t)c * HW_];
        m[c] = x[c] + (1.0f - slab[b*C_+c]) * NEG_;
        mx = fmaxf(mx, m[c]);
    }
    float se = 0.0f;
    #pragma unroll
    for (int c = 0; c < C_; ++c) { m[c] = expf(m[c] - mx); se += m[c]; }
    float inv = 1.0f / se;
    #pragma unroll
    for (int c = 0; c < C_; ++c)
        out2[((size_t)(b*C_+c) * HW_ + hw) * V_ + v] = m[c] * inv;
    #pragma unroll
    for (int c = 0; c < C_; ++c) atomicAdd(&psum[b*C_+c], x[c]);
    __syncthreads();
    if (tid < B_*C_) atomicAdd(&ws[84+tid], psum[tid]);
}

// ----------------------------------------------------- margin losses ----
__device__ __forceinline__ float logsigf(float x) {
    return fminf(x, 0.0f) - log1pf(expf(-fabsf(x)));
}

__global__ void margin_kernel(const float* __restrict__ lab3,
                              const float* __restrict__ lab2,
                              const float* __restrict__ ws,
                              float* __restrict__ loss_out)
{
    __shared__ float red[128];
    int tid = threadIdx.x;
    float val = 0.0f;
    if (tid < B_*C_) {
        int b = tid / C_;
        float x3 = ws[tid] / ws[80 + b];
        float y3 = lab3[tid];
        float t3 = y3 * logsigf(x3) + (1.0f - y3) * logsigf(-x3);
        float x2 = ws[84 + tid] * (1.0f / (float)(HW_ * V_));
        float y2 = lab2[tid];
        float t2 = y2 * logsigf(x2) + (1.0f - y2) * logsigf(-x2);
        val = -(t3 + t2) * (1.0f / (float)(B_*C_));
    }
    red[tid] = val;
    __syncthreads();
    for (int s = 64; s > 0; s >>= 1) {
        if (tid < s) red[tid] += red[tid + s];
        __syncthreads();
    }
    if (tid == 0) atomicAdd(loss_out, red[0]);
}

// ------------------------------------------------- CMG losses (WMMA) ----
// 512 blocks = 2 losses * 4 batches * 64 row-tiles; 256 threads = 8 waves.
// A-matrix (16 K-rows) lives in registers (16 x v2f per lane, ISA f32 layout:
// lane lo = M, hi selects K-pair). Q tiles are staged per-wave into
// double-buffered LDS with async global->LDS copies overlapped via ASYNCcnt.
// Online (max,sum) logsumexp: single WMMA pass over the 1024 columns.
// Diagonal element of row M lives in lane with lo==M (r = lo-8*hi), captured
// into a register under the uniform isdiag branch; one LDS store at the end.
__global__ __launch_bounds__(256) void cmg_kernel(
    const float* __restrict__ k3, const float* __restrict__ q3,
    const float* __restrict__ k2, const float* __restrict__ q2,
    const int* __restrict__ idx3, const int* __restrict__ idx2,
    float* __restrict__ loss_out)
{
    int lb     = blockIdx.x;
    int rowblk = lb & 63;
    int b      = (lb >> 6) & 3;
    int lossid = lb >> 8;
    const float* fk  = lossid ? k2   : k3;
    const float* fq  = lossid ? q2   : q3;
    const int*   idx = lossid ? idx2 : idx3;

    __shared__ float qbuf[8][2][16][68];   // per-wave double-buffered Q tile (padded)
    __shared__ float maxpart[8][16];
    __shared__ float sumpart[8][16];
    __shared__ float diagv[16];

    int tid  = threadIdx.x;
    int wave = tid >> 5;
    int lane = tid & 31;
    int hi   = lane >> 4;     // K-pair selector (ISA f32 A/B layout)
    int lo   = lane & 15;     // M (A) / N (B) index
    int i0   = rowblk * 16;

    // ---- A matrix in registers: lane lo's K-row, the 2*hi K-pair of each k0 ----
    v2f areg[16];
    {
        int src = idx[b * K_ + i0 + lo];
        const float* arow = fk + ((size_t)b * P_ + src) * D_ + 2 * hi;
        #pragma unroll
        for (int j = 0; j < 16; ++j) areg[j] = *(const v2f*)(arow + 4 * j);
    }

    const float inv_tau = 1.0f / TAU_;
    int wd = rowblk & 7, jd = rowblk >> 3;   // wave/iter owning the diagonal tile
    int rdiag = lo - 8 * hi;                 // acc slot holding this lane's diagonal (if 0..7)

    float mxa[8], sma[8];
    float dreg = 0.0f;
    #pragma unroll
    for (int r = 0; r < 8; ++r) { mxa[r] = -INFINITY; sma[r] = 0.0f; }

    // ---- stage helper (wave-private tile: 16 rows x 64 floats) ----
    auto stage_q = [&](int m0, float* dst) {
        int r = lane >> 1, half = lane & 1;
        int src = idx[b * K_ + m0 + r];
        const float* gp = fq + ((size_t)b * P_ + src) * D_ + half * 32;
        float* lp = dst + r * 68 + half * 32;
#if defined(ASYNC_LDS)
        #pragma unroll
        for (int jj = 0; jj < 8; ++jj)
            __builtin_amdgcn_global_load_async_to_lds_b128(G1(gp + jj*4), L3(lp + jj*4), 0, 0);
#else
        #pragma unroll
        for (int jj = 0; jj < 8; ++jj) {
            float4 vv = *(const float4*)(gp + jj*4);
            *(float4*)(lp + jj*4) = vv;
        }
#endif
    };

    // prologue: stage tile for j=0 into buffer 0
    stage_q(wave * 16, &qbuf[wave][0][0][0]);

    for (int j = 0; j < 8; ++j) {
        int buf = j & 1;
        if (j < 7)
            stage_q((wave + 8*(j+1)) * 16, &qbuf[wave][buf ^ 1][0][0]);
        if (j < 7) { ASYNC_WAIT(8); } else { ASYNC_WAIT(0); }  // prev tile landed
        asm volatile("" ::: "memory");

        v8f acc = {};
        #pragma unroll
        for (int k0 = 0; k0 < 64; k0 += 4) {
            v2f bb = *(const v2f*)&qbuf[wave][buf][lo][k0 + 2*hi];
            acc = __builtin_amdgcn_wmma_f32_16x16x4_f32(
                      false, areg[k0 >> 2], false, bb, (short)0, acc, false, false);
        }

        if ((wave == wd) && (j == jd)) {     // uniform per wave: register select only
            #pragma unroll
            for (int r = 0; r < 8; ++r)
                if (r == rdiag) dreg = acc[r] * inv_tau;
        }
        #pragma unroll
        for (int r = 0; r < 8; ++r) {
            float s = acc[r] * inv_tau;
            float mn = fmaxf(mxa[r], s);
            sma[r] = sma[r] * expf(mxa[r] - mn) + expf(s - mn);
            mxa[r] = mn;
        }
    }

    // single predicated diagonal store (rows M == lo; lanes 0-7 and 24-31)
    if ((wave == wd) && (rdiag >= 0) && (rdiag < 8)) diagv[lo] = dreg;

    // ---- combine (m,s) across the 16 lanes of each half-wave (rows r+8*hi) ----
    #pragma unroll
    for (int r = 0; r < 8; ++r) {
        float m = mxa[r], s = sma[r];
        #pragma unroll
        for (int msk = 8; msk >= 1; msk >>= 1) {
            float m2 = __shfl_xor(m, msk, 32);
            float s2 = __shfl_xor(s, msk, 32);
            float mn = fmaxf(m, m2);
            s = s * expf(m - mn) + s2 * expf(m2 - mn);
            m = mn;
        }
        if (lo == 0) { maxpart[wave][hi*8 + r] = m; sumpart[wave][hi*8 + r] = s; }
    }
    __syncthreads();
    if (tid < 16) {
        float m = maxpart[0][tid], s = sumpart[0][tid];
        for (int w = 1; w < 8; ++w) {
            float m2 = maxpart[w][tid], s2 = sumpart[w][tid];
            float mn = fmaxf(m, m2);
            s = s * expf(m - mn) + s2 * expf(m2 - mn);
            m = mn;
        }
        float lse = m + logf(s);
        atomicAdd(loss_out, (lse - diagv[tid]) * (1.0f / (float)K_));
    }
}

// ---------------------------------------------------------------- host ----
extern "C" void kernel_launch(void* const* d_in, const int* in_sizes, int n_in,
                              void* d_out, int out_size, void* d_ws, size_t ws_size,
                              hipStream_t stream) {
    const float* cam3d = (const float*)d_in[0];
    const float* cam2d = (const float*)d_in[1];
    const float* fp3   = (const float*)d_in[2];
    const float* fd2   = (const float*)d_in[3];
    const float* fp2   = (const float*)d_in[4];
    const float* fd3   = (const float*)d_in[5];
    const float* lab3  = (const float*)d_in[6];
    const float* lab2  = (const float*)d_in[7];
    const int*   invm  = (const int*)d_in[8];
    const int*   bidx  = (const int*)d_in[9];
    const int*   mi3   = (const int*)d_in[10];
    const int*   mi2   = (const int*)d_in[11];

    float* out = (float*)d_out;                  // [0]=loss, [1..]=result_3d, then result_2d
    float* ws  = (float*)d_ws;
    float* out3 = out + 1;
    float* out2 = out + 1 + (size_t)N_ * C_;

    init_kernel<<<1, 256, 0, stream>>>(ws, out);
    cam3d_kernel<<<(N_ + 255) / 256, 256, 0, stream>>>(cam3d, invm, bidx, lab3, out3, ws);
    cam2d_kernel<<<(B_ * V_ * HW_) / 256, 256, 0, stream>>>(cam2d, lab2, out2, ws);
    margin_kernel<<<1, 128, 0, stream>>>(lab3, lab2, ws, out);
    cmg_kernel<<<512, 256, 0, stream>>>(fp3, fd2, fp2, fd3, mi3, mi2, out);
}